// RoIHeads_48713519072064
// MI455X (gfx1250) — compile-verified
//
#include <hip/hip_runtime.h>
#include <math.h>

#define B_    8
#define N_    8192
#define C_    80
#define NC_   (N_ * C_)          // 655360 per image
#define NBINS 4096
#define CAP   2048               // candidate buffer per image
#define KPAD  1024               // padded NMS set
#define KPRE  1000               // reference K_PRE
#define DET   100
#define SCORE_T 0.05f
#define MIN_SZ  0.01f
#define XCLIP   4.135166556742356f   // log(1000/16)

typedef float v2f __attribute__((ext_vector_type(2)));
typedef float v8f __attribute__((ext_vector_type(8)));

// ---------------------------------------------------------------- decode ----
__device__ __forceinline__ void decode_one(
    const float* __restrict__ logits, const float* __restrict__ regs,
    const float* __restrict__ props, int b, int n, int c, float Hf, float Wf,
    float& score, float& x1, float& y1, float& x2, float& y2, bool& valid)
{
    int bn = b * N_ + n;
    float lg = logits[(size_t)bn * C_ + c];
    score = 1.0f / (1.0f + __expf(-lg));
    float4 code = ((const float4*)regs)[(size_t)bn * C_ + c];
    float4 p    = ((const float4*)props)[bn];
    float w = p.z - p.x, h = p.w - p.y;
    float cx = p.x + 0.5f * w, cy = p.y + 0.5f * h;
    float dx = code.x * 0.1f, dy = code.y * 0.1f;
    float dw = fminf(code.z * 0.2f, XCLIP), dh = fminf(code.w * 0.2f, XCLIP);
    float pcx = dx * w + cx, pcy = dy * h + cy;
    float pw = __expf(dw) * w, ph = __expf(dh) * h;
    x1 = fminf(fmaxf(pcx - 0.5f * pw, 0.f), Wf);
    y1 = fminf(fmaxf(pcy - 0.5f * ph, 0.f), Hf);
    x2 = fminf(fmaxf(pcx + 0.5f * pw, 0.f), Wf);
    y2 = fminf(fmaxf(pcy + 0.5f * ph, 0.f), Hf);
    valid = (score > SCORE_T) && ((x2 - x1) >= MIN_SZ) && ((y2 - y1) >= MIN_SZ);
}

// ------------------------------------------------------------------ init ----
__global__ void k_init(unsigned* hist, unsigned* counter, unsigned* maxc, int* thr)
{
    int gid = blockIdx.x * 256 + threadIdx.x;
    if (gid < B_ * NBINS) hist[gid] = 0u;
    if (gid < B_) { counter[gid] = 0u; maxc[gid] = 0u; thr[gid] = 0; }
}

// --------------------------------------------- pass 1: histogram + maxbox ---
__global__ void k_hist(const float* __restrict__ logits, const float* __restrict__ regs,
                       const float* __restrict__ props, const int* ihp, const int* iwp,
                       unsigned* __restrict__ hist, unsigned* __restrict__ maxc)
{
    int gid = blockIdx.x * blockDim.x + threadIdx.x;
    int b = gid / NC_;
    int r = gid - b * NC_;
    int n = r / C_;
    int c = r - n * C_;
    float Hf = (float)(*ihp), Wf = (float)(*iwp);
    float s, x1, y1, x2, y2; bool valid;
    decode_one(logits, regs, props, b, n, c, Hf, Wf, s, x1, y1, x2, y2, valid);
    if (valid) {
        int bin = (int)(s * (float)NBINS);
        bin = min(bin, NBINS - 1);
        atomicAdd(&hist[b * NBINS + bin], 1u);
    }
    // jnp.max(flat_boxes): over ALL clipped boxes (coords >= 0 -> uint order ok)
    unsigned m = __float_as_uint(fmaxf(fmaxf(x1, y1), fmaxf(x2, y2)));
    __shared__ unsigned red[256];
    red[threadIdx.x] = m;
    __syncthreads();
    for (int st = 128; st > 0; st >>= 1) {
        if (threadIdx.x < st) red[threadIdx.x] = max(red[threadIdx.x], red[threadIdx.x + st]);
        __syncthreads();
    }
    if (threadIdx.x == 0) atomicMax(&maxc[b], red[0]);
}

// ---------------------------------- find score-bin threshold for top-KPRE ---
__global__ void k_thresh(const unsigned* __restrict__ hist, int* __restrict__ thr)
{
    int b = blockIdx.x;
    int lane = threadIdx.x;        // one wave32 per image
    unsigned s = 0;
    int p0 = lane * 128;           // positions from highest bin downward
    for (int q = 0; q < 128; ++q) s += hist[b * NBINS + (NBINS - 1 - (p0 + q))];
    unsigned v = s;
    for (int d = 1; d < 32; d <<= 1) {
        unsigned t = __shfl_up(v, d, 32);
        if (lane >= d) v += t;
    }
    unsigned pre = v - s;          // exclusive prefix (higher-score strips)
    if (pre < KPRE && pre + s >= KPRE) {
        unsigned cum = pre;
        for (int q = 0; q < 128; ++q) {
            int bin = NBINS - 1 - (p0 + q);
            cum += hist[b * NBINS + bin];
            if (cum >= KPRE) { thr[b] = bin; break; }
        }
    }
    unsigned total = __shfl(v, 31, 32);
    if (lane == 31 && total < KPRE) thr[b] = 0;
}

// ------------------------------------------- pass 2: compact candidates -----
__global__ void k_compact(const float* __restrict__ logits, const float* __restrict__ regs,
                          const float* __restrict__ props, const int* ihp, const int* iwp,
                          const int* __restrict__ thr, unsigned* __restrict__ counter,
                          float* __restrict__ cand_s, unsigned* __restrict__ cand_i,
                          float* __restrict__ cand_box)
{
    int gid = blockIdx.x * blockDim.x + threadIdx.x;
    int b = gid / NC_;
    int r = gid - b * NC_;
    int n = r / C_;
    int c = r - n * C_;
    float Hf = (float)(*ihp), Wf = (float)(*iwp);
    float s, x1, y1, x2, y2; bool valid;
    decode_one(logits, regs, props, b, n, c, Hf, Wf, s, x1, y1, x2, y2, valid);
    if (valid) {
        int bin = min((int)(s * (float)NBINS), NBINS - 1);
        if (bin >= thr[b]) {
            unsigned pos = atomicAdd(&counter[b], 1u);
            if (pos < CAP) {
                cand_s[b * CAP + pos] = s;
                cand_i[b * CAP + pos] = (unsigned)(n * C_ + c);  // flat idx: label = idx % C
                float4 bx; bx.x = x1; bx.y = y1; bx.z = x2; bx.w = y2;
                ((float4*)cand_box)[b * CAP + pos] = bx;
            }
        }
    }
}

// ------------------------------- bitonic sort candidates, emit sorted top ---
__global__ void __launch_bounds__(1024)
k_sort(const float* __restrict__ cand_s, const unsigned* __restrict__ cand_i,
       const float* __restrict__ cand_box, const unsigned* __restrict__ counter,
       float* __restrict__ sscore, float* __restrict__ sbox, int* __restrict__ slabel)
{
    int b = blockIdx.x;
    int tid = threadIdx.x;
    __shared__ float    ks[CAP];
    __shared__ unsigned ki[CAP];
    __shared__ unsigned kslot[CAP];
    unsigned cnt = counter[b]; if (cnt > CAP) cnt = CAP;
    for (int t = tid; t < CAP; t += 1024) {
        if ((unsigned)t < cnt) { ks[t] = cand_s[b * CAP + t]; ki[t] = cand_i[b * CAP + t]; kslot[t] = (unsigned)t; }
        else                   { ks[t] = -1e30f; ki[t] = 0xFFFFFFFFu; kslot[t] = 0xFFFFFFFFu; }
    }
    for (unsigned sz = 2; sz <= CAP; sz <<= 1)
        for (unsigned j = sz >> 1; j > 0; j >>= 1) {
            __syncthreads();
            for (unsigned t = tid; t < CAP; t += 1024) {
                unsigned l = t ^ j;
                if (l > t) {
                    bool blockFirst = ((t & sz) == 0);
                    float sa = ks[t], sb = ks[l];
                    unsigned ia = ki[t], ib = ki[l];
                    bool aFirst = (sa > sb) || (sa == sb && ia < ib);  // desc score, asc idx
                    if (blockFirst ? !aFirst : aFirst) {
                        ks[t] = sb; ks[l] = sa;
                        ki[t] = ib; ki[l] = ia;
                        unsigned tmp = kslot[t]; kslot[t] = kslot[l]; kslot[l] = tmp;
                    }
                }
            }
        }
    __syncthreads();
    int r = tid;               // 0..1023
    if (r < KPRE && (unsigned)r < cnt) {
        sscore[b * KPAD + r] = ks[r];
        ((float4*)sbox)[b * KPAD + r] = ((const float4*)cand_box)[b * CAP + kslot[r]];
        slabel[b * KPAD + r] = (int)(ki[r] % C_);
    } else {
        sscore[b * KPAD + r] = -1.f;
        float4 z; z.x = z.y = z.z = z.w = 0.f;
        ((float4*)sbox)[b * KPAD + r] = z;
        slabel[b * KPAD + r] = 0;
    }
}

// ---------------- suppression bit-matrix: one 16x16 tile per wave, WMMA -----
// S[i][j] = area_i + area_j via V_WMMA_F32_16X16X4_F32 (rank-2 matmul);
// iou > 0.5  <=>  3*inter > S  (union = S - inter, divide-free).
__global__ void __launch_bounds__(256)
k_bitmat(const float* __restrict__ sbox, const int* __restrict__ slabel,
         const unsigned* __restrict__ maxc, unsigned short* __restrict__ bitmat)
{
    int wave = threadIdx.x >> 5;
    int lane = threadIdx.x & 31;
    int gtile = blockIdx.x * 8 + wave;         // B_*64*64 = 32768 tiles total
    int b  = gtile >> 12;                      // /4096
    int t  = gtile & 4095;
    int mt = t >> 6, nt = t & 63;

    __shared__ float sA[8][16][5];             // x1,y1,x2,y2,area (offset boxes)
    __shared__ float sB[8][16][5];

    float maxcf = __uint_as_float(maxc[b]);
    if (lane < 16) {
        int row = mt * 16 + lane;
        float4 bx = ((const float4*)sbox)[b * KPAD + row];
        float ofs = (float)slabel[b * KPAD + row] * (maxcf + 1.0f);
        sA[wave][lane][0] = bx.x + ofs; sA[wave][lane][1] = bx.y + ofs;
        sA[wave][lane][2] = bx.z + ofs; sA[wave][lane][3] = bx.w + ofs;
        sA[wave][lane][4] = (bx.z - bx.x) * (bx.w - bx.y);
    } else {
        int col = nt * 16 + (lane - 16);
        float4 bx = ((const float4*)sbox)[b * KPAD + col];
        float ofs = (float)slabel[b * KPAD + col] * (maxcf + 1.0f);
        sB[wave][lane - 16][0] = bx.x + ofs; sB[wave][lane - 16][1] = bx.y + ofs;
        sB[wave][lane - 16][2] = bx.z + ofs; sB[wave][lane - 16][3] = bx.w + ofs;
        sB[wave][lane - 16][4] = (bx.z - bx.x) * (bx.w - bx.y);
    }
    __syncthreads();

    bool lo = lane < 16;
    // A (16x4): K0 = areas, K1 = ones; lanes 16..31 carry K2/K3 = 0
    v2f A;  A.x  = lo ? sA[wave][lane][4] : 0.f;  A.y  = lo ? 1.f : 0.f;
    // B (4x16): K0 = ones, K1 = areas
    v2f Bv; Bv.x = lo ? 1.f : 0.f;                Bv.y = lo ? sB[wave][lane][4] : 0.f;
    v8f acc = {};
    acc = __builtin_amdgcn_wmma_f32_16x16x4_f32(false, A, false, Bv,
                                                (short)0, acc, false, false);

    int rbase = (b * KPAD + mt * 16) * 64 + nt;  // u16 chunks, row pitch 64
    int Nc = lane & 15;
    float bx1 = sB[wave][Nc][0], by1 = sB[wave][Nc][1];
    float bx2 = sB[wave][Nc][2], by2 = sB[wave][Nc][3];
#pragma unroll
    for (int k = 0; k < 8; ++k) {
        int M = lo ? k : (k + 8);                // matches C/D VGPR layout
        float ax1 = sA[wave][M][0], ay1 = sA[wave][M][1];
        float ax2 = sA[wave][M][2], ay2 = sA[wave][M][3];
        float iw = fmaxf(fminf(ax2, bx2) - fmaxf(ax1, bx1), 0.f);
        float ih = fmaxf(fminf(ay2, by2) - fmaxf(ay1, by1), 0.f);
        float inter = iw * ih;
        bool sup = (3.0f * inter > acc[k]);      // IoU > 0.5
        unsigned msk = __builtin_amdgcn_ballot_w32(sup);
        if (lane == k)      bitmat[rbase + k * 64]       = (unsigned short)(msk & 0xFFFFu);
        if (lane == 16 + k) bitmat[rbase + (k + 8) * 64] = (unsigned short)(msk >> 16);
    }
}

// ----------------- serial greedy NMS on one wave + ranked top-100 output ----
__global__ void __launch_bounds__(1024)
k_final(const float* __restrict__ sscore, const float* __restrict__ sbox,
        const int* __restrict__ slabel, const unsigned short* __restrict__ bitmat,
        float* __restrict__ out)
{
    int b = blockIdx.x;
    int tid = threadIdx.x;
    __shared__ unsigned keepw[32];
    __shared__ unsigned char alivef[KPAD];
    alivef[tid] = (sscore[b * KPAD + tid] > 0.f) ? 1 : 0;
    __syncthreads();

    if (tid < 32) {                              // one wave32 holds 1024 keep bits
        unsigned keep = 0;
        for (int q = 0; q < 32; ++q) keep |= ((unsigned)alivef[tid * 32 + q]) << q;
        for (int i = 0; i < KPRE; ++i) {
            unsigned kw = (unsigned)__shfl((int)keep, i >> 5, 32);
            if ((kw >> (i & 31)) & 1u) {
                const unsigned* row = (const unsigned*)(bitmat + (size_t)(b * KPAD + i) * 64);
                unsigned rv = row[tid];
                int base = tid * 32;
                unsigned msk;
                if (i < base)             msk = 0xFFFFFFFFu;
                else if (i >= base + 31)  msk = 0u;
                else                      msk = 0xFFFFFFFEu << (i - base);  // cols j > i
                keep &= ~(rv & msk);
            }
        }
        keepw[tid] = keep;
    }
    __syncthreads();

    if (tid < DET) {                             // default rows: zeros, label -1
        float* o = out + ((size_t)b * DET + tid) * 6;
        o[0] = 0.f; o[1] = 0.f; o[2] = 0.f; o[3] = 0.f; o[4] = 0.f; o[5] = -1.f;
    }
    __syncthreads();

    unsigned kb = (keepw[tid >> 5] >> (tid & 31)) & 1u;
    if (kb) {
        int w = tid >> 5;
        int rank = 0;
        for (int q = 0; q < w; ++q) rank += __popc(keepw[q]);
        rank += __popc(keepw[w] & ((1u << (tid & 31)) - 1u));
        if (rank < DET) {                        // sorted order => kept order = top-k
            float4 bx = ((const float4*)sbox)[b * KPAD + tid];
            float* o = out + ((size_t)b * DET + rank) * 6;
            o[0] = bx.x; o[1] = bx.y; o[2] = bx.z; o[3] = bx.w;
            o[4] = sscore[b * KPAD + tid];
            o[5] = (float)slabel[b * KPAD + tid];
        }
    }
}

// ------------------------------------------------------------------ launch --
extern "C" void kernel_launch(void* const* d_in, const int* in_sizes, int n_in,
                              void* d_out, int out_size, void* d_ws, size_t ws_size,
                              hipStream_t stream)
{
    const float* logits = (const float*)d_in[0];
    const float* regs   = (const float*)d_in[1];
    const float* props  = (const float*)d_in[2];
    const int*   ihp    = (const int*)d_in[3];
    const int*   iwp    = (const int*)d_in[4];
    float* out = (float*)d_out;

    // workspace carve-up (~1.8 MB total; float4 regions 16B-aligned)
    unsigned* hist    = (unsigned*)d_ws;                       // B*4096
    unsigned* counter = hist + B_ * NBINS;                     // B
    unsigned* maxc    = counter + B_;                          // B
    int*      thr     = (int*)(maxc + B_);                     // B
    float*    cand_s  = (float*)(thr + B_);                    // B*CAP
    unsigned* cand_i  = (unsigned*)(cand_s + B_ * CAP);        // B*CAP
    float*    cand_box= (float*)(cand_i + B_ * CAP);           // B*CAP*4
    float*    sscore  = cand_box + (size_t)B_ * CAP * 4;       // B*KPAD
    float*    sbox    = sscore + B_ * KPAD;                    // B*KPAD*4
    int*      slabel  = (int*)(sbox + (size_t)B_ * KPAD * 4);  // B*KPAD
    unsigned short* bitmat = (unsigned short*)(slabel + B_ * KPAD); // B*KPAD*64 u16

    k_init   <<<(B_ * NBINS + 255) / 256, 256, 0, stream>>>(hist, counter, maxc, thr);
    k_hist   <<<(B_ * NC_) / 256, 256, 0, stream>>>(logits, regs, props, ihp, iwp, hist, maxc);
    k_thresh <<<B_, 32, 0, stream>>>(hist, thr);
    k_compact<<<(B_ * NC_) / 256, 256, 0, stream>>>(logits, regs, props, ihp, iwp, thr,
                                                    counter, cand_s, cand_i, cand_box);
    k_sort   <<<B_, 1024, 0, stream>>>(cand_s, cand_i, cand_box, counter, sscore, sbox, slabel);
    k_bitmat <<<(B_ * 64 * 64) / 8, 256, 0, stream>>>(sbox, slabel, maxc, bitmat);
    k_final  <<<B_, 1024, 0, stream>>>(sscore, sbox, slabel, bitmat, out);
}